// RoIBBox_21930103014209
// MI455X (gfx1250) — compile-verified
//
#include <hip/hip_runtime.h>
#include <stdint.h>

#define BATCH   16
#define A_TOTAL 147456
#define TOPK    1000
#define PADK    1024
#define CAP     4096
#define IOU_THR 0.7f

typedef __attribute__((ext_vector_type(2))) float v2f;
typedef __attribute__((ext_vector_type(8))) float v8f;

__device__ __forceinline__ unsigned mkey(float f) {
  unsigned u = __float_as_uint(f);
  return (u & 0x80000000u) ? ~u : (u | 0x80000000u);
}

// ---------------- init workspace (deterministic per call) ----------------
__global__ void init_ws(unsigned long long* __restrict__ key64,
                        unsigned* __restrict__ counter) {
  int t = blockIdx.x * blockDim.x + threadIdx.x;
  if (t < BATCH * CAP) key64[t] = 0ull;
  if (t < BATCH) counter[t] = 0u;
}

// ---------------- per-row softmax stats: max + sum(exp) ----------------
__global__ void softmax_stats(const float* __restrict__ labels,
                              float* __restrict__ row_max,
                              float* __restrict__ row_sum) {
  __shared__ float red[1024];
  int b = blockIdx.x, tid = threadIdx.x;
  const float* row = labels + (size_t)b * A_TOTAL;
  float lm = -3.4e38f;
  for (int i = tid; i < A_TOTAL; i += 1024) {
    __builtin_prefetch(row + i + 4096, 0, 0);   // global_prefetch_b8
    lm = fmaxf(lm, row[i]);
  }
  red[tid] = lm; __syncthreads();
  for (int s = 512; s > 0; s >>= 1) {
    if (tid < s) red[tid] = fmaxf(red[tid], red[tid + s]);
    __syncthreads();
  }
  float rmax = red[0]; __syncthreads();
  float ls = 0.f;
  for (int i = tid; i < A_TOTAL; i += 1024) ls += expf(row[i] - rmax);
  red[tid] = ls; __syncthreads();
  for (int s = 512; s > 0; s >>= 1) {
    if (tid < s) red[tid] += red[tid + s];
    __syncthreads();
  }
  if (tid == 0) { row_max[b] = rmax; row_sum[b] = red[0]; }
}

// ---------------- MSB radix select: exact 1000th-largest key ----------------
__global__ void radix_select(const float* __restrict__ labels,
                             unsigned* __restrict__ thresh) {
  __shared__ unsigned hist[256];
  __shared__ unsigned s_prefix;
  __shared__ int s_k;
  int b = blockIdx.x, tid = threadIdx.x, bs = blockDim.x;
  const float* row = labels + (size_t)b * A_TOTAL;
  unsigned prefix = 0u; int k = TOPK;
  for (int digit = 3; digit >= 0; --digit) {
    for (int i = tid; i < 256; i += bs) hist[i] = 0u;
    __syncthreads();
    unsigned pm = (digit == 3) ? 0u : (0xFFFFFFFFu << ((digit + 1) * 8));
    for (int i = tid; i < A_TOTAL; i += bs) {
      __builtin_prefetch(row + i + 4096, 0, 0);
      unsigned m = mkey(row[i]);
      if ((m & pm) == (prefix & pm))
        atomicAdd(&hist[(m >> (digit * 8)) & 0xFFu], 1u);
    }
    __syncthreads();
    if (tid == 0) {
      int acc = 0, sel = 0;
      for (int bin = 255; bin >= 0; --bin) {
        acc += (int)hist[bin];
        if (acc >= k) { sel = bin; break; }
      }
      s_k = k - (acc - (int)hist[sel]);
      s_prefix = prefix | ((unsigned)sel << (digit * 8));
    }
    __syncthreads();
    k = s_k; prefix = s_prefix;
    __syncthreads();
  }
  if (tid == 0) thresh[b] = prefix;
}

// ---------------- gather candidates >= threshold ----------------
__global__ void compact(const float* __restrict__ labels,
                        const unsigned* __restrict__ thresh,
                        unsigned* __restrict__ counter,
                        unsigned long long* __restrict__ key64) {
  int t = blockIdx.x * blockDim.x + threadIdx.x;  // exactly BATCH*A_TOTAL
  int b = t / A_TOTAL, i = t % A_TOTAL;
  unsigned m = mkey(labels[(size_t)b * A_TOTAL + i]);
  if (m >= thresh[b]) {
    unsigned pos = atomicAdd(&counter[b], 1u);
    if (pos < CAP)
      key64[b * CAP + pos] =
          ((unsigned long long)m << 32) | (unsigned)(~(unsigned)i);
  }
}

// ---------------- per-batch bitonic sort (desc) -> top indices ----------------
__global__ void sort_topk(const unsigned long long* __restrict__ key64,
                          unsigned* __restrict__ top_idx) {
  __shared__ unsigned long long s[CAP];
  int b = blockIdx.x, tid = threadIdx.x;
  for (int i = tid; i < CAP; i += 1024) s[i] = key64[b * CAP + i];
  __syncthreads();
  for (unsigned k = 2; k <= CAP; k <<= 1) {
    for (unsigned j = k >> 1; j > 0; j >>= 1) {
      for (unsigned i = tid; i < CAP; i += 1024) {
        unsigned ixj = i ^ j;
        if (ixj > i) {
          unsigned long long a = s[i], c = s[ixj];
          bool up = ((i & k) == 0);
          if (up ? (a < c) : (a > c)) { s[i] = c; s[ixj] = a; }
        }
      }
      __syncthreads();
    }
  }
  for (int r = tid; r < PADK; r += 1024) {
    unsigned idx = ~(unsigned)(s[r] & 0xFFFFFFFFull);
    top_idx[b * PADK + r] = idx;
  }
}

// ---------------- decode only the selected boxes + final scores ----------------
__global__ void decode_topk(const float* __restrict__ deltas,
                            const float* __restrict__ labels,
                            const float* __restrict__ anchors,
                            const float* __restrict__ variances,
                            const unsigned* __restrict__ top_idx,
                            const float* __restrict__ row_max,
                            const float* __restrict__ row_sum,
                            float4* __restrict__ boxes,
                            float* __restrict__ scores) {
  int t = blockIdx.x * blockDim.x + threadIdx.x;  // BATCH*PADK
  int b = t >> 10, r = t & 1023;
  float4 box = make_float4(0.f, 0.f, 0.f, 0.f);
  float sc = 0.f;
  unsigned idx = top_idx[b * PADK + r];
  if (r < TOPK && idx < (unsigned)A_TOTAL) {
    float logit = labels[(size_t)b * A_TOTAL + idx];
    sc = expf(logit - row_max[b]) / row_sum[b];
    const float* dp = deltas + ((size_t)b * A_TOTAL + idx) * 4;
    float d0 = dp[0] * variances[0], d1 = dp[1] * variances[1];
    float d2 = dp[2] * variances[2], d3 = dp[3] * variances[3];
    float a0 = anchors[idx * 4 + 0], a1 = anchors[idx * 4 + 1];
    float a2 = anchors[idx * 4 + 2], a3 = anchors[idx * 4 + 3];
    float aw = a3 - a1, ah = a2 - a0;
    float acx = a1 + 0.5f * aw, acy = a0 + 0.5f * ah;
    float bw = expf(d3) * aw, bh = expf(d2) * ah;
    float bcx = d1 * aw + acx, bcy = d0 * ah + acy;
    float y1 = bcy - 0.5f * bh, x1 = bcx - 0.5f * bw;
    float y2 = bh + y1, x2 = bw + x1;
    box = make_float4(y1, x1, y2, x2);
  }
  boxes[b * PADK + r] = box;
  scores[b * PADK + r] = sc;
}

// ---------------- pairwise IoU > thr bitmask; WMMA computes area_i+area_j ----------------
__global__ void __launch_bounds__(256) iou_mask(const float4* __restrict__ boxes,
                                                unsigned* __restrict__ mask) {
  __shared__ float4 ldsI[8][16];
  __shared__ float4 ldsJ[8][32];
  int wv = threadIdx.x >> 5, lane = threadIdx.x & 31;
  int wg = blockIdx.x * 8 + wv;          // 16 batches * 64 row-tiles * 32 col-words
  int b = wg >> 11;
  int rem = wg & 2047;
  int it = rem >> 5, jw = rem & 31;
  int i0 = it << 4, j0 = jw << 5;

  if (lane < 16) ldsI[wv][lane] = boxes[b * PADK + i0 + lane];
  ldsJ[wv][lane] = boxes[b * PADK + j0 + lane];
  __syncthreads();

  // A fragment (16x4 f32): row m -> [area_i[m], 1, 0, 0]
  v2f a; a.x = 0.f; a.y = 0.f;
  if (lane < 16) {
    float4 bi = ldsI[wv][lane];
    a.x = (bi.z - bi.x) * (bi.w - bi.y);
    a.y = 1.0f;
  }

  unsigned bal[2][8];
#pragma unroll
  for (int t = 0; t < 2; ++t) {
    // B fragment (4x16 f32): col n -> [1, area_j[n], 0, 0]^T
    v2f bf; bf.x = 0.f; bf.y = 0.f;
    if (lane < 16) {
      float4 bj = ldsJ[wv][t * 16 + lane];
      bf.x = 1.0f;
      bf.y = (bj.z - bj.x) * (bj.w - bj.y);
    }
    v8f c = {0.f, 0.f, 0.f, 0.f, 0.f, 0.f, 0.f, 0.f};
    // D[m][n] = area_i[m] + area_j[n]  (union precursor via rank-2 matmul)
    c = __builtin_amdgcn_wmma_f32_16x16x4_f32(false, a, false, bf,
                                              (short)0, c, false, false);
    int m_base = (lane >> 4) << 3;  // lanes 0-15: M=v; lanes 16-31: M=v+8
    int n = lane & 15;
    float4 bj = ldsJ[wv][t * 16 + n];
#pragma unroll
    for (int v = 0; v < 8; ++v) {
      float4 bi = ldsI[wv][m_base + v];
      float ih = fminf(bi.z, bj.z) - fmaxf(bi.x, bj.x);
      float iw = fminf(bi.w, bj.w) - fmaxf(bi.y, bj.y);
      float inter = fmaxf(ih, 0.f) * fmaxf(iw, 0.f);
      float uni = c[v] - inter;
      // inter/max(uni,eps) > THR  <=>  inter > THR*max(uni,eps)   (both >= 0)
      bool pred = inter > IOU_THR * fmaxf(uni, 1e-9f);
      bal[t][v] = (unsigned)__ballot(pred);
    }
  }
#pragma unroll
  for (int v = 0; v < 8; ++v) {
    unsigned b0 = bal[0][v], b1 = bal[1][v];
    unsigned wlow  = (b0 & 0xFFFFu) | ((b1 & 0xFFFFu) << 16);  // row i0+v
    unsigned whigh = (b0 >> 16) | (b1 & 0xFFFF0000u);          // row i0+v+8
    if (lane == 0) mask[((size_t)b * PADK + i0 + v) * 32 + jw] = wlow;
    if (lane == 1) mask[((size_t)b * PADK + i0 + v + 8) * 32 + jw] = whigh;
  }
}

// ---------------- sequential NMS scan over the bitmask (1 wave / batch) ----------------
__global__ void nms_scan(const unsigned* __restrict__ mask,
                         unsigned* __restrict__ keepW) {
  int b = blockIdx.x, lane = threadIdx.x;  // 32 threads
  unsigned sup = 0u, keep = 0u;
  for (int i = 0; i < TOPK; ++i) {
    int wi = i >> 5, bi = i & 31;
    unsigned supw = __shfl(sup, wi, 32);
    if (((supw >> bi) & 1u) == 0u) {
      unsigned row = mask[((size_t)b * PADK + i) * 32 + lane];
      if (lane < wi) row = 0u;
      else if (lane == wi) row &= (bi == 31) ? 0u : ~((1u << (bi + 1)) - 1u);
      sup |= row;
      if (lane == wi) keep |= (1u << bi);
    }
  }
  keepW[b * 32 + lane] = keep;
}

// ---------------- stable kept-first compaction + zero fill ----------------
__global__ void write_out(const float4* __restrict__ boxes,
                          const float* __restrict__ scores,
                          const unsigned* __restrict__ keepW,
                          float* __restrict__ out) {
  __shared__ unsigned kw[32];
  int b = blockIdx.x, tid = threadIdx.x;
  if (tid < 32) kw[tid] = keepW[b * 32 + tid];
  __syncthreads();
  if (tid < TOPK) {
    out[(b * TOPK + tid) * 4 + 0] = 0.f;
    out[(b * TOPK + tid) * 4 + 1] = 0.f;
    out[(b * TOPK + tid) * 4 + 2] = 0.f;
    out[(b * TOPK + tid) * 4 + 3] = 0.f;
    out[BATCH * TOPK * 4 + b * TOPK + tid] = 0.f;
  }
  __syncthreads();
  if (tid < TOPK) {
    int w = tid >> 5, bi = tid & 31;
    if ((kw[w] >> bi) & 1u) {
      int rank = 0;
      for (int q = 0; q < w; ++q) rank += __popc(kw[q]);
      rank += __popc(kw[w] & ((bi == 0) ? 0u : ((1u << bi) - 1u)));
      float4 bx = boxes[b * PADK + tid];
      out[(b * TOPK + rank) * 4 + 0] = bx.x;
      out[(b * TOPK + rank) * 4 + 1] = bx.y;
      out[(b * TOPK + rank) * 4 + 2] = bx.z;
      out[(b * TOPK + rank) * 4 + 3] = bx.w;
      out[BATCH * TOPK * 4 + b * TOPK + rank] = scores[b * PADK + tid];
    }
  }
}

extern "C" void kernel_launch(void* const* d_in, const int* in_sizes, int n_in,
                              void* d_out, int out_size, void* d_ws, size_t ws_size,
                              hipStream_t stream) {
  const float* deltas    = (const float*)d_in[0];
  const float* labels    = (const float*)d_in[1];
  const float* anchors   = (const float*)d_in[2];
  const float* variances = (const float*)d_in[3];
  float* out = (float*)d_out;

  char* w = (char*)d_ws;
  float*              row_max = (float*)(w + 0);
  float*              row_sum = (float*)(w + 64);
  unsigned*           thresh  = (unsigned*)(w + 128);
  unsigned*           counter = (unsigned*)(w + 192);
  unsigned long long* key64   = (unsigned long long*)(w + 256);       // 16*4096*8 = 524288
  unsigned*           top_idx = (unsigned*)(w + 524544);              // 16*1024*4 = 65536
  float4*             boxes   = (float4*)(w + 590080);                // 16*1024*16 = 262144
  float*              scores  = (float*)(w + 852224);                 // 16*1024*4 = 65536
  unsigned*           mask    = (unsigned*)(w + 917760);              // 16*1024*32*4 = 2097152
  unsigned*           keepW   = (unsigned*)(w + 3014912);             // 16*32*4 = 2048

  init_ws<<<256, 256, 0, stream>>>(key64, counter);
  softmax_stats<<<BATCH, 1024, 0, stream>>>(labels, row_max, row_sum);
  radix_select<<<BATCH, 1024, 0, stream>>>(labels, thresh);
  compact<<<(BATCH * A_TOTAL) / 256, 256, 0, stream>>>(labels, thresh, counter, key64);
  sort_topk<<<BATCH, 1024, 0, stream>>>(key64, top_idx);
  decode_topk<<<(BATCH * PADK) / 256, 256, 0, stream>>>(
      deltas, labels, anchors, variances, top_idx, row_max, row_sum, boxes, scores);
  iou_mask<<<(BATCH * 64 * 32) / 8, 256, 0, stream>>>(boxes, mask);
  nms_scan<<<BATCH, 32, 0, stream>>>(mask, keepW);
  write_out<<<BATCH, 1024, 0, stream>>>(boxes, scores, keepW, out);
}